// simpel_model_70360154243520
// MI455X (gfx1250) — compile-verified
//
#include <hip/hip_runtime.h>

typedef __attribute__((ext_vector_type(2))) float v2f;
typedef __attribute__((ext_vector_type(8))) float v8f;

#define NB 1024
#define NK 2048
#define ND 32
#define NL 10000

// ---------------------------------------------------------------------------
// Phase 1: c0 = x @ (y + z)   [1024 x 32]
// One wave (32 lanes) per output row; lane = output column d.
// Memory-bound and tiny (134 MFLOP); x streamed once, y/z L2-resident.
// ---------------------------------------------------------------------------
__global__ __launch_bounds__(256) void c0_kernel(const float* __restrict__ x,
                                                 const float* __restrict__ y,
                                                 const float* __restrict__ z,
                                                 float* __restrict__ c0) {
    const int d = threadIdx.x;                        // 0..31
    const int row = blockIdx.x * 8 + threadIdx.y;     // 8 rows per block
    const float* xr = x + (size_t)row * NK;

    float acc = 0.0f;
#pragma unroll 1
    for (int k = 0; k < NK; k += 8) {
        float4 x0 = *(const float4*)(xr + k);
        float4 x1 = *(const float4*)(xr + k + 4);
        acc = fmaf(x0.x, y[(k + 0) * ND + d] + z[(k + 0) * ND + d], acc);
        acc = fmaf(x0.y, y[(k + 1) * ND + d] + z[(k + 1) * ND + d], acc);
        acc = fmaf(x0.z, y[(k + 2) * ND + d] + z[(k + 2) * ND + d], acc);
        acc = fmaf(x0.w, y[(k + 3) * ND + d] + z[(k + 3) * ND + d], acc);
        acc = fmaf(x1.x, y[(k + 4) * ND + d] + z[(k + 4) * ND + d], acc);
        acc = fmaf(x1.y, y[(k + 5) * ND + d] + z[(k + 5) * ND + d], acc);
        acc = fmaf(x1.z, y[(k + 6) * ND + d] + z[(k + 6) * ND + d], acc);
        acc = fmaf(x1.w, y[(k + 7) * ND + d] + z[(k + 7) * ND + d], acc);
    }
    c0[(size_t)row * ND + d] = acc;
}

// ---------------------------------------------------------------------------
// Phase 2: 10000 sequential Linear(32,32) layers via V_WMMA_F32_16X16X4_F32.
// One wave owns a 16-row tile of c. Per layer:
//   - 16 WMMAs (8 k-steps x 2 N-tiles) seeded with a hoisted zero C,
//   - 2 WMMAs for the rank-1 bias update (consumes the prefetched bias LAST),
//   - LDS round-trip (c^T col-major) for D-layout -> A-fragment relayout.
// No __syncthreads(): single-wave workgroup + in-order LDS per wave means only
// a compiler fence is needed, so the layer i+1 W/bias prefetch stays in flight
// across the whole layer (true double buffering).
// ---------------------------------------------------------------------------
__global__ __launch_bounds__(32) void chain_kernel(const float* __restrict__ W,
                                                   const float* __restrict__ bias,
                                                   float* __restrict__ cio) {
    __shared__ __align__(32) float tile[16 * ND];   // c^T col-major: tile[col*16 + row]

    const int lane  = threadIdx.x;   // 0..31
    const int lcol  = lane & 15;
    const int lhalf = lane >> 4;
    const int row0  = blockIdx.x * 16;

    // Loop-invariant helpers (hoisted): zero C seed, and the constant A fragment
    // for the rank-1 bias update: A[m][0] = 1, A[m][k>0] = 0  =>  D += bias row.
    const v8f zero8 = {0.f, 0.f, 0.f, 0.f, 0.f, 0.f, 0.f, 0.f};
    const v2f abias = {lhalf ? 0.0f : 1.0f, 0.0f};

    // ---- initial A fragments straight from c0 (in cio) ----
    // A 16x4 f32 layout: lanes 0-15 hold K={k0,k0+1}, lanes 16-31 hold K={k0+2,k0+3}
    v2f afrag[8];
    {
        const float* cp = cio + (size_t)(row0 + lcol) * ND + 2 * lhalf;
#pragma unroll
        for (int kk = 0; kk < 8; ++kk)
            afrag[kk] = *(const v2f*)(cp + kk * 4);
    }

    // ---- preload layer 0 weights/bias (B = W^T, so B[k][n] = W[n*32 + k]) ----
    v2f  wbuf[2][2][8];
    float bbuf[2][2];
    bbuf[0][0] = bias[lcol];
    bbuf[0][1] = bias[16 + lcol];
#pragma unroll
    for (int t = 0; t < 2; ++t) {
        const float* wp = W + (size_t)(t * 16 + lcol) * ND + 2 * lhalf;
#pragma unroll
        for (int kk = 0; kk < 8; ++kk)
            wbuf[0][t][kk] = *(const v2f*)(wp + kk * 4);
    }

#pragma unroll 2
    for (int i = 0; i < NL; ++i) {
        const int cur = i & 1;
        const int nxt = cur ^ 1;

        // ---- prefetch layer i+1 (bias first: in-order returns => cheap waits) ----
        if (i + 1 < NL) {
            const float* Wn = W + (size_t)(i + 1) * (ND * ND);
            const float* bn = bias + (size_t)(i + 1) * ND;
            bbuf[nxt][0] = bn[lcol];
            bbuf[nxt][1] = bn[16 + lcol];
#pragma unroll
            for (int t = 0; t < 2; ++t) {
                const float* wp = Wn + (size_t)(t * 16 + lcol) * ND + 2 * lhalf;
#pragma unroll
                for (int kk = 0; kk < 8; ++kk)
                    wbuf[nxt][t][kk] = *(const v2f*)(wp + kk * 4);
            }
        }

        // ---- c_new = c @ W^T : 8 accumulate k-steps per N-tile, C seeded 0 ----
        v8f acc0 = zero8;
        v8f acc1 = zero8;
#pragma unroll
        for (int kk = 0; kk < 8; ++kk) {
            acc0 = __builtin_amdgcn_wmma_f32_16x16x4_f32(
                false, afrag[kk], false, wbuf[cur][0][kk], (short)0, acc0, false, false);
            acc1 = __builtin_amdgcn_wmma_f32_16x16x4_f32(
                false, afrag[kk], false, wbuf[cur][1][kk], (short)0, acc1, false, false);
        }

        // ---- rank-1 bias update on the matrix pipe (bias consumed last) ----
        // B fragment: K=0 row holds bias[n] (lanes 0-15), all other K rows zero.
        {
            v2f bb0 = {lhalf ? 0.0f : bbuf[cur][0], 0.0f};
            v2f bb1 = {lhalf ? 0.0f : bbuf[cur][1], 0.0f};
            acc0 = __builtin_amdgcn_wmma_f32_16x16x4_f32(
                false, abias, false, bb0, (short)0, acc0, false, false);
            acc1 = __builtin_amdgcn_wmma_f32_16x16x4_f32(
                false, abias, false, bb1, (short)0, acc1, false, false);
        }

        if (i + 1 < NL) {
            // ---- D-layout -> LDS (c^T col-major) -> reload as A fragments ----
            // D tile: VGPR v, lane half h -> row v+8h, col lcol.
            *(v8f*)&tile[(0 * 16 + lcol) * 16 + 8 * lhalf] = acc0;  // 2x ds_store_b128
            *(v8f*)&tile[(1 * 16 + lcol) * 16 + 8 * lhalf] = acc1;
            asm volatile("" ::: "memory");  // compiler fence; LDS is in-order per wave
#pragma unroll
            for (int kk = 0; kk < 8; ++kk) {
                const int base = (kk * 4 + 2 * lhalf) * 16 + lcol;
                v2f a;
                a[0] = tile[base];        // c[row=lcol][k0 + 2*lhalf]
                a[1] = tile[base + 16];   // c[row=lcol][k0 + 2*lhalf + 1]
                afrag[kk] = a;
            }
            asm volatile("" ::: "memory");  // keep next store after these loads
        } else {
            // ---- final layer: scatter D-layout to output ----
            float* op = cio + (size_t)(row0 + 8 * lhalf) * ND + lcol;
#pragma unroll
            for (int v = 0; v < 8; ++v) {
                op[(size_t)v * ND + 0]  = acc0[v];
                op[(size_t)v * ND + 16] = acc1[v];
            }
        }
    }
}

// ---------------------------------------------------------------------------
extern "C" void kernel_launch(void* const* d_in, const int* in_sizes, int n_in,
                              void* d_out, int out_size, void* d_ws, size_t ws_size,
                              hipStream_t stream) {
    const float* x = (const float*)d_in[0];   // [1024, 2048]
    const float* y = (const float*)d_in[1];   // [2048, 32]
    const float* z = (const float*)d_in[2];   // [2048, 32]
    const float* W = (const float*)d_in[3];   // [10000, 32, 32]
    const float* b = (const float*)d_in[4];   // [10000, 32]
    float* out = (float*)d_out;               // [1024, 32] — also used as c0 staging

    (void)in_sizes; (void)n_in; (void)out_size; (void)d_ws; (void)ws_size;

    c0_kernel<<<dim3(NB / 8), dim3(32, 8), 0, stream>>>(x, y, z, out);
    chain_kernel<<<dim3(NB / 16), dim3(32), 0, stream>>>(W, b, out);
}